// RN_B_72267119722519
// MI455X (gfx1250) — compile-verified
//
#include <hip/hip_runtime.h>

#define B_ 32
#define C_ 256
#define L_ 8192
#define NPER (B_ * L_)          // elements per channel = 262144
#define BN_EPS 1e-5f

typedef float v2f __attribute__((ext_vector_type(2)));
typedef float v4f __attribute__((ext_vector_type(4)));
typedef float v8f __attribute__((ext_vector_type(8)));

// workspace layout (floats)
#define WS_PS   0                    // [C_*B_] partial sum   (c*B_+b)
#define WS_PQ   (C_*B_)              // [C_*B_] partial sumsq
#define WS_PMS  (2*C_*B_)            // [B_] masked partial sum (channel 0)
#define WS_PMQ  (2*C_*B_ + B_)       // [B_] masked partial sumsq
#define WS_A1   (2*C_*B_ + 2*B_)     // [C_] affine (mask==1) scale
#define WS_B1   (WS_A1 + C_)         // [C_] affine (mask==1) bias
#define WS_A0   (WS_A1 + 2*C_)       // [C_] affine (mask==0) scale
#define WS_B0   (WS_A1 + 3*C_)       // [C_] affine (mask==0) bias

// ---------------------------------------------------------------------------
// Pass 1: per-(channel, batch) partial sums via WMMA f32 16x16x4 accumulation.
// Each block: one (c, b) pair, 256 threads, L_ = 8192 contiguous floats.
// With B-matrix == all-ones: D[m,n] = sum_k A[m,k] + C[m,n]  (layout-agnostic
// reduction; all 16 columns hold identical totals -> divide lane sum by 16).
// Loads are RT (default) so the tail of x stays resident in the 192MB L2 for
// the apply pass, which traverses in reverse order.
// ---------------------------------------------------------------------------
__global__ __launch_bounds__(256) void stats_kernel(const float* __restrict__ x,
                                                    float* __restrict__ ws) {
    const int c   = blockIdx.x >> 5;
    const int b   = blockIdx.x & 31;
    const int tid = threadIdx.x;
    const float* __restrict__ xp = x + ((size_t)b * C_ + (size_t)c) * L_;

    v8f acc_s = {};
    v8f acc_q = {};
    const v2f ones = {1.0f, 1.0f};

#pragma unroll
    for (int i = 0; i < 8; ++i) {
        v4f v = *(const v4f*)(xp + i * 1024 + tid * 4);   // RT: keep in L2
        v2f a0 = {v.x, v.y};
        v2f a1 = {v.z, v.w};
        v2f q0 = {v.x * v.x, v.y * v.y};
        v2f q1 = {v.z * v.z, v.w * v.w};
        acc_s = __builtin_amdgcn_wmma_f32_16x16x4_f32(false, a0, false, ones,
                                                      (short)0, acc_s, false, false);
        acc_s = __builtin_amdgcn_wmma_f32_16x16x4_f32(false, a1, false, ones,
                                                      (short)0, acc_s, false, false);
        acc_q = __builtin_amdgcn_wmma_f32_16x16x4_f32(false, q0, false, ones,
                                                      (short)0, acc_q, false, false);
        acc_q = __builtin_amdgcn_wmma_f32_16x16x4_f32(false, q1, false, ones,
                                                      (short)0, acc_q, false, false);
    }

    float s = 0.0f, q = 0.0f;
#pragma unroll
    for (int k = 0; k < 8; ++k) { s += acc_s[k]; q += acc_q[k]; }

    __shared__ float rs[256];
    __shared__ float rq[256];
    rs[tid] = s; rq[tid] = q;
    __syncthreads();
    for (int off = 128; off > 0; off >>= 1) {
        if (tid < off) { rs[tid] += rs[tid + off]; rq[tid] += rq[tid + off]; }
        __syncthreads();
    }
    if (tid == 0) {
        // every one of the 16 N-columns of D holds the full total
        ws[WS_PS + c * B_ + b] = rs[0] * 0.0625f;
        ws[WS_PQ + c * B_ + b] = rq[0] * 0.0625f;
    }
}

// ---------------------------------------------------------------------------
// Pass 1b: masked sums over channel 0 only (tiny: 1 MiB of x + 1 MiB of mask).
// ---------------------------------------------------------------------------
__global__ __launch_bounds__(256) void mstats_kernel(const float* __restrict__ x,
                                                     const float* __restrict__ mask,
                                                     float* __restrict__ ws) {
    const int b   = blockIdx.x;
    const int tid = threadIdx.x;
    const float* __restrict__ xp = x + (size_t)b * C_ * L_;   // channel 0 row
    const float* __restrict__ mp = mask + (size_t)b * L_;

    float sm = 0.0f, qm = 0.0f;
#pragma unroll
    for (int i = 0; i < 8; ++i) {
        v4f v = *(const v4f*)(xp + i * 1024 + tid * 4);
        v4f m = *(const v4f*)(mp + i * 1024 + tid * 4);
        v4f xm = v * m;                       // m in {0,1}
        sm += xm.x + xm.y + xm.z + xm.w;
        qm += xm.x * v.x + xm.y * v.y + xm.z * v.z + xm.w * v.w;  // x^2 * m
    }

    __shared__ float rs[256];
    __shared__ float rq[256];
    rs[tid] = sm; rq[tid] = qm;
    __syncthreads();
    for (int off = 128; off > 0; off >>= 1) {
        if (tid < off) { rs[tid] += rs[tid + off]; rq[tid] += rq[tid + off]; }
        __syncthreads();
    }
    if (tid == 0) {
        ws[WS_PMS + b] = rs[0];
        ws[WS_PMQ + b] = rq[0];
    }
}

__device__ __forceinline__ float rsqrt_refined(float v) {
    float r = rsqrtf(v);
    r = r * (1.5f - 0.5f * v * r * r);   // one Newton step -> full f32 accuracy
    return r;
}

// ---------------------------------------------------------------------------
// Pass 2: fold BN stats + affine params into 4 per-channel scalars.
// Single block, thread c handles channel c (deterministic sequential sums).
// ---------------------------------------------------------------------------
__global__ __launch_bounds__(256) void coef_kernel(const float* __restrict__ fg_gamma,
                                                   const float* __restrict__ fg_beta,
                                                   const float* __restrict__ bg_gamma,
                                                   const float* __restrict__ bg_beta,
                                                   float* __restrict__ ws) {
    const int c = threadIdx.x;
    float s = 0.0f, q = 0.0f;
    for (int b = 0; b < B_; ++b) {
        s += ws[WS_PS + c * B_ + b];
        q += ws[WS_PQ + c * B_ + b];
    }
    const float invN = 1.0f / (float)NPER;
    const float fgg = fg_gamma[c], fgb = fg_beta[c];
    const float bgg = bg_gamma[c], bgb = bg_beta[c];

    float a1, b1, a0, b0;
    if (c == 0) {
        float sm = 0.0f, qm = 0.0f;
        for (int b = 0; b < B_; ++b) { sm += ws[WS_PMS + b]; qm += ws[WS_PMQ + b]; }
        const float mu1  = sm * invN;
        const float var1 = qm * invN - mu1 * mu1;
        const float r1   = rsqrt_refined(var1 + BN_EPS);
        const float mu2  = (s - sm) * invN;
        const float var2 = (q - qm) * invN - mu2 * mu2;
        const float r2   = rsqrt_refined(var2 + BN_EPS);
        const float K    = -(mu1 * r1 + mu2 * r2);
        a1 = r1 * (1.0f + fgg);
        b1 = K  * (1.0f + fgg) + fgb + bgb;
        a0 = r2 * (1.0f + bgg);
        b0 = K  * (1.0f + bgg) + fgb + bgb;
    } else {
        // mask1 == 0 for c >= 1: BN(0)=0, rn = BN(x); pure affine, mask-agnostic
        const float mu  = s * invN;
        const float var = q * invN - mu * mu;
        const float r   = rsqrt_refined(var + BN_EPS);
        const float a   = r * (1.0f + bgg);
        const float bb  = -mu * a + fgb + bgb;
        a1 = a; a0 = a; b1 = bb; b0 = bb;
    }
    ws[WS_A1 + c] = a1;
    ws[WS_B1 + c] = b1;
    ws[WS_A0 + c] = a0;
    ws[WS_B0 + c] = b0;
}

// ---------------------------------------------------------------------------
// Pass 3: out = m ? x*a1[c]+b1[c] : x*a0[c]+b0[c].  float4 per thread.
// Block index is REVERSED so this pass consumes x from the tail (still L2-
// resident from pass 1) backwards -> real reuse instead of LRU thrash.
// NT load of x (last use), NT store of out (write-once): keep mask + x hot.
// Block -> single (b, c, 1024-float chunk): coeff reads are wave-uniform.
// ---------------------------------------------------------------------------
__global__ __launch_bounds__(256) void apply_kernel(const float* __restrict__ x,
                                                    const float* __restrict__ mask,
                                                    const float* __restrict__ ws,
                                                    float* __restrict__ out) {
    const int blk   = (B_ * C_ * 8 - 1) - blockIdx.x;   // reverse traversal
    const int chunk = blk & 7;
    const int c     = (blk >> 3) & (C_ - 1);
    const int b     = blk >> 11;
    const int l     = chunk * 1024 + threadIdx.x * 4;

    const float a1 = ws[WS_A1 + c], b1 = ws[WS_B1 + c];
    const float a0 = ws[WS_A0 + c], b0 = ws[WS_B0 + c];

    const size_t xoff = ((size_t)b * C_ + (size_t)c) * L_ + (size_t)l;
    v4f v = __builtin_nontemporal_load((const v4f*)(x + xoff));
    v4f m = *(const v4f*)(mask + (size_t)b * L_ + (size_t)l);

    v4f r;
    r.x = (m.x > 0.5f) ? (v.x * a1 + b1) : (v.x * a0 + b0);
    r.y = (m.y > 0.5f) ? (v.y * a1 + b1) : (v.y * a0 + b0);
    r.z = (m.z > 0.5f) ? (v.z * a1 + b1) : (v.z * a0 + b0);
    r.w = (m.w > 0.5f) ? (v.w * a1 + b1) : (v.w * a0 + b0);

    __builtin_nontemporal_store(r, (v4f*)(out + xoff));
}

extern "C" void kernel_launch(void* const* d_in, const int* in_sizes, int n_in,
                              void* d_out, int out_size, void* d_ws, size_t ws_size,
                              hipStream_t stream) {
    (void)in_sizes; (void)n_in; (void)out_size; (void)ws_size;
    const float* x    = (const float*)d_in[0];
    const float* mask = (const float*)d_in[1];
    const float* fgg  = (const float*)d_in[2];
    const float* fgb  = (const float*)d_in[3];
    const float* bgg  = (const float*)d_in[4];
    const float* bgb  = (const float*)d_in[5];
    float* ws  = (float*)d_ws;
    float* out = (float*)d_out;

    stats_kernel<<<C_ * B_, 256, 0, stream>>>(x, ws);          // 8192 blocks
    mstats_kernel<<<B_, 256, 0, stream>>>(x, mask, ws);        // 32 blocks
    coef_kernel<<<1, 256, 0, stream>>>(fgg, fgb, bgg, bgb, ws);
    apply_kernel<<<B_ * C_ * 8, 256, 0, stream>>>(x, mask, ws, out); // 65536 blocks
}